// Mamba2_83167746720235
// MI455X (gfx1250) — compile-verified
//
#include <hip/hip_runtime.h>
#include <hip/hip_bf16.h>
#include <math.h>

#define B_      128
#define HIDDEN_ 2048
#define INNER_  4096
#define HEAD_   64
#define NH_     64
#define BC_     128
#define EPS_    1e-5f

typedef float v2f __attribute__((ext_vector_type(2)));
typedef float v8f __attribute__((ext_vector_type(8)));

// --- CDNA5 async direct-to-LDS copy (ASYNCcnt-tracked, bypasses VGPRs) ------
// Generic shared-pointer low 32 bits == LDS byte offset (flat LDS aperture).
__device__ __forceinline__ uint32_t lds_off(const void* p) {
    return (uint32_t)(uintptr_t)p;
}
__device__ __forceinline__ void async_ld_b128(uint32_t dst_lds, const void* src) {
    asm volatile("global_load_async_to_lds_b128 %0, %1, off"
                 :: "v"(dst_lds), "v"(src) : "memory");
}
// Async loads complete in-order: waiting <=6 releases the previous chunk's
// 6 ops/wave even while the next chunk's 6 are still in flight.
__device__ __forceinline__ void wait_async_le6() {
    asm volatile("s_wait_asynccnt 0x6" ::: "memory");
}
__device__ __forceinline__ void wait_async_0() {
    asm volatile("s_wait_asynccnt 0x0" ::: "memory");
}

// ---------------------------------------------------------------------------
// C[128 x N] = A[128 x K] @ W[N x K]^T (+ optional per-column bias)
// fp32 exact math via V_WMMA_F32_16X16X4_F32.
// grid.x = N/64, block = 256 threads (8 waves). Block tile: M=128, N=64.
// K staged through double-buffered LDS chunks of 32 via async-to-LDS DMA;
// rows padded to 36 floats (16B-aligned B128 stores, conflict-free b64 reads).
// ---------------------------------------------------------------------------
__device__ __forceinline__ void stage_chunk(const float* __restrict__ A,
                                            const float* __restrict__ W,
                                            int K, int nblk, int t, int k0,
                                            float (*As)[36], float (*Ws)[36]) {
#pragma unroll
    for (int r = 0; r < 4; ++r) {               // A chunk: 128 x 32
        const int row = (t >> 3) + r * 32;
        const int c4  = (t & 7) * 4;
        async_ld_b128(lds_off(&As[row][c4]), A + (size_t)row * K + k0 + c4);
    }
#pragma unroll
    for (int r = 0; r < 2; ++r) {               // W chunk: 64 x 32
        const int row = (t >> 3) + r * 32;
        const int c4  = (t & 7) * 4;
        async_ld_b128(lds_off(&Ws[row][c4]),
                      W + (size_t)(nblk + row) * K + k0 + c4);
    }
}

__global__ __launch_bounds__(256)
void gemm_f32_wmma(const float* __restrict__ A, const float* __restrict__ W,
                   const float* __restrict__ bias, float* __restrict__ C,
                   int K, int N) {
    __shared__ float As[2][128][36];
    __shared__ float Ws[2][64][36];

    const int t    = threadIdx.x;
    const int wave = t >> 5;
    const int lane = t & 31;
    const int wn   = wave & 3;   // n sub-tile within block (0..3)
    const int wm2  = wave >> 2;  // 0/1 -> rows [0,64) or [64,128)
    const int nblk = blockIdx.x * 64;

    const int lcol  = lane & 15;        // N (B-frag) or M (A-frag) index
    const int khalf = (lane >> 4) * 2;  // lanes 16-31 hold K+2,K+3

    v8f acc[4];
#pragma unroll
    for (int i = 0; i < 4; ++i)
        acc[i] = v8f{0.f, 0.f, 0.f, 0.f, 0.f, 0.f, 0.f, 0.f};

    const int nchunks = K >> 5;
    stage_chunk(A, W, K, nblk, t, 0, As[0], Ws[0]);

    for (int i = 0; i < nchunks; ++i) {
        const int cur = i & 1;
        if (i + 1 < nchunks) {
            // overlap next chunk's DMA with this chunk's WMMA work
            stage_chunk(A, W, K, nblk, t, (i + 1) * 32, As[cur ^ 1], Ws[cur ^ 1]);
            wait_async_le6();          // current buffer's 6 ops/wave done
        } else {
            wait_async_0();
        }
        __syncthreads();               // all waves' LDS writes visible

#pragma unroll
        for (int k = 0; k < 32; k += 4) {
            // B frag (4x16): VGPR0=K(khalf), VGPR1=K(khalf+1), N = lane&15
            const v2f bfrag = *(const v2f*)&Ws[cur][wn * 16 + lcol][k + khalf];
#pragma unroll
            for (int j = 0; j < 4; ++j) {
                const int mrow = wm2 * 64 + j * 16 + lcol;
                const v2f afrag = *(const v2f*)&As[cur][mrow][k + khalf];
                acc[j] = __builtin_amdgcn_wmma_f32_16x16x4_f32(
                    false, afrag, false, bfrag, (short)0, acc[j], false, false);
            }
        }
        __syncthreads();               // done reading before buffer reuse
    }

    // Epilogue: VGPR r -> (M = mbase + r + 8*(lane>=16), N = nbase + (lane&15))
    const int half8 = (lane >> 4) * 8;
    const int col   = nblk + wn * 16 + lcol;
    const float bv  = bias ? bias[col] : 0.f;
#pragma unroll
    for (int i = 0; i < 4; ++i) {
        const int mbase = wm2 * 64 + i * 16;
#pragma unroll
        for (int r = 0; r < 8; ++r) {
            const int row = mbase + r + half8;
            C[(size_t)row * N + col] = acc[i][r] + bv;
        }
    }
}

// ---------------------------------------------------------------------------
// Depthwise causal conv step (K=4) + SiLU; also emits new conv state.
// state layout [B, C, 3]; new state = {s1, s2, x_raw}
// ---------------------------------------------------------------------------
__global__ void conv_silu_kernel(const float* __restrict__ pre,
                                 const float* __restrict__ state_in,
                                 const float* __restrict__ w,
                                 const float* __restrict__ bias,
                                 float* __restrict__ act,
                                 float* __restrict__ state_out, int C) {
    const int idx = blockIdx.x * blockDim.x + threadIdx.x;
    if (idx >= B_ * C) return;
    const int ch = idx % C;
    const size_t sb = (size_t)idx * 3;
    const float s0 = state_in[sb + 0];
    const float s1 = state_in[sb + 1];
    const float s2 = state_in[sb + 2];
    const float xr = pre[idx];
    const float v = w[ch * 4 + 0] * s0 + w[ch * 4 + 1] * s1 +
                    w[ch * 4 + 2] * s2 + w[ch * 4 + 3] * xr + bias[ch];
    act[idx] = v / (1.f + __expf(-v));     // SiLU
    state_out[sb + 0] = s1;
    state_out[sb + 1] = s2;
    state_out[sb + 2] = xr;
}

// dt = softplus(raw); decay = exp(a[n]*dt)
__global__ void dt_kernel(const float* __restrict__ raw,
                          const float* __restrict__ a,
                          float* __restrict__ dtsp,
                          float* __restrict__ decay) {
    const int idx = blockIdx.x * blockDim.x + threadIdx.x;
    if (idx >= B_ * NH_) return;
    const int n = idx & (NH_ - 1);
    const float r  = raw[idx];
    const float sp = (r > 20.f) ? r : log1pf(__expf(r));
    dtsp[idx]  = sp;
    decay[idx] = __expf(a[n] * sp);
}

// ---------------------------------------------------------------------------
// SSM state update + y contraction. One block per (b, n); 64x128 fp32 tile.
// Bandwidth-dominant: ~536 MB state traffic -> sets the ~26us roofline floor.
// ---------------------------------------------------------------------------
__global__ __launch_bounds__(256)
void ssm_update_kernel(const float* __restrict__ state,
                       const float* __restrict__ x_act,
                       const float* __restrict__ b_act,
                       const float* __restrict__ c_act,
                       const float* __restrict__ dtsp,
                       const float* __restrict__ decay,
                       const float* __restrict__ d,
                       float* __restrict__ state_new,
                       float* __restrict__ y) {
    const int blk = blockIdx.x;      // b*NH + n
    const int b   = blk >> 6;
    const int n   = blk & 63;
    const int t   = threadIdx.x;
    const int h   = t >> 2;
    const int c0  = (t & 3) * 32;

    const float dec = decay[blk];
    const float dtv = dtsp[blk];
    const float xv  = x_act[(size_t)b * INNER_ + n * HEAD_ + h];
    const float kx  = dtv * xv;

    const float* bp = b_act + (size_t)b * BC_ + c0;
    const float* cp = c_act + (size_t)b * BC_ + c0;
    const size_t base = (size_t)blk * HEAD_ * BC_ + (size_t)h * BC_ + c0;
    const float* sp = state + base;
    float* np = state_new + base;

    float acc = 0.f;
#pragma unroll
    for (int j = 0; j < 32; j += 4) {
        const float4 s  = *(const float4*)(sp + j);
        const float4 bb = *(const float4*)(bp + j);
        const float4 cc = *(const float4*)(cp + j);
        float4 nv;
        nv.x = dec * s.x + kx * bb.x;
        nv.y = dec * s.y + kx * bb.y;
        nv.z = dec * s.z + kx * bb.z;
        nv.w = dec * s.w + kx * bb.w;
        acc += cc.x * nv.x + cc.y * nv.y + cc.z * nv.z + cc.w * nv.w;
        *(float4*)(np + j) = nv;
    }
    acc += __shfl_xor(acc, 1);
    acc += __shfl_xor(acc, 2);
    if ((t & 3) == 0)
        y[(size_t)b * INNER_ + n * HEAD_ + h] = acc + d[n] * xv;
}

// ---------------------------------------------------------------------------
// y = y * silu(z); RMSNorm over 4096; scale by norm_w. One block per batch row.
// ---------------------------------------------------------------------------
__global__ __launch_bounds__(256)
void gate_norm_kernel(const float* __restrict__ y, const float* __restrict__ z,
                      const float* __restrict__ norm_w, float* __restrict__ yn) {
    const int b = blockIdx.x;
    const int t = threadIdx.x;
    __shared__ float red[8];

    float g[16];
    float ss = 0.f;
#pragma unroll
    for (int j = 0; j < 16; ++j) {
        const int col = t + j * 256;
        const float yv = y[(size_t)b * INNER_ + col];
        const float zv = z[(size_t)b * INNER_ + col];
        const float gz = zv / (1.f + __expf(-zv));
        const float gg = yv * gz;
        g[j] = gg;
        ss += gg * gg;
    }
#pragma unroll
    for (int off = 16; off >= 1; off >>= 1) ss += __shfl_xor(ss, off);
    if ((t & 31) == 0) red[t >> 5] = ss;
    __syncthreads();
    if (t == 0) {
        float s = 0.f;
#pragma unroll
        for (int w = 0; w < 8; ++w) s += red[w];
        red[0] = rsqrtf(s / (float)INNER_ + EPS_);
    }
    __syncthreads();
    const float r = red[0];
#pragma unroll
    for (int j = 0; j < 16; ++j) {
        const int col = t + j * 256;
        yn[(size_t)b * INNER_ + col] = g[j] * r * norm_w[col];
    }
}

// ---------------------------------------------------------------------------
extern "C" void kernel_launch(void* const* d_in, const int* in_sizes, int n_in,
                              void* d_out, int out_size, void* d_ws, size_t ws_size,
                              hipStream_t stream) {
    const float* tin   = (const float*)d_in[0];
    const float* x_cs0 = (const float*)d_in[1];
    const float* b_cs0 = (const float*)d_in[2];
    const float* c_cs0 = (const float*)d_in[3];
    const float* ssm0  = (const float*)d_in[4];
    const float* W_in  = (const float*)d_in[5];
    const float* W_z   = (const float*)d_in[6];
    const float* W_b   = (const float*)d_in[7];
    const float* W_c   = (const float*)d_in[8];
    const float* W_dt  = (const float*)d_in[9];
    const float* b_dt  = (const float*)d_in[10];
    const float* cxw   = (const float*)d_in[11];
    const float* cxb   = (const float*)d_in[12];
    const float* cbw   = (const float*)d_in[13];
    const float* cbb   = (const float*)d_in[14];
    const float* ccw   = (const float*)d_in[15];
    const float* ccb   = (const float*)d_in[16];
    const float* a_p   = (const float*)d_in[17];
    const float* d_p   = (const float*)d_in[18];
    const float* nw    = (const float*)d_in[19];
    const float* W_out = (const float*)d_in[20];

    // Workspace carve-up (floats)
    float* ws = (float*)d_ws;
    size_t o = 0;
    float* x_pre  = ws + o; o += (size_t)B_ * INNER_;
    float* z_pre  = ws + o; o += (size_t)B_ * INNER_;
    float* b_pre  = ws + o; o += (size_t)B_ * BC_;
    float* c_pre  = ws + o; o += (size_t)B_ * BC_;
    float* dt_raw = ws + o; o += (size_t)B_ * NH_;
    float* x_act  = ws + o; o += (size_t)B_ * INNER_;
    float* b_act  = ws + o; o += (size_t)B_ * BC_;
    float* c_act  = ws + o; o += (size_t)B_ * BC_;
    float* dtsp   = ws + o; o += (size_t)B_ * NH_;
    float* decay  = ws + o; o += (size_t)B_ * NH_;
    float* y_buf  = ws + o; o += (size_t)B_ * INNER_;
    float* yn_buf = ws + o; o += (size_t)B_ * INNER_;

    // Output carve-up: (out, x_cs, b_cs, c_cs, ssm_new) flat-concatenated
    float* out     = (float*)d_out;
    float* x_cs    = out  + (size_t)B_ * HIDDEN_;
    float* b_cs    = x_cs + (size_t)B_ * INNER_ * 3;
    float* c_cs    = b_cs + (size_t)B_ * BC_ * 3;
    float* ssm_new = c_cs + (size_t)B_ * BC_ * 3;

    // 1) fp32 WMMA projections (weights all L2-resident: ~103 MB < 192 MB)
    gemm_f32_wmma<<<INNER_ / 64, 256, 0, stream>>>(tin, W_in, nullptr, x_pre, HIDDEN_, INNER_);
    gemm_f32_wmma<<<INNER_ / 64, 256, 0, stream>>>(tin, W_z,  nullptr, z_pre, HIDDEN_, INNER_);
    gemm_f32_wmma<<<BC_ / 64,    256, 0, stream>>>(tin, W_b,  nullptr, b_pre, HIDDEN_, BC_);
    gemm_f32_wmma<<<BC_ / 64,    256, 0, stream>>>(tin, W_c,  nullptr, c_pre, HIDDEN_, BC_);
    gemm_f32_wmma<<<NH_ / 64,    256, 0, stream>>>(tin, W_dt, b_dt,  dt_raw, HIDDEN_, NH_);

    // 2) Conv steps + SiLU (+ new conv states straight into d_out)
    conv_silu_kernel<<<(B_ * INNER_) / 256, 256, 0, stream>>>(x_pre, x_cs0, cxw, cxb, x_act, x_cs, INNER_);
    conv_silu_kernel<<<(B_ * BC_) / 256,    256, 0, stream>>>(b_pre, b_cs0, cbw, cbb, b_act, b_cs, BC_);
    conv_silu_kernel<<<(B_ * BC_) / 256,    256, 0, stream>>>(c_pre, c_cs0, ccw, ccb, c_act, c_cs, BC_);

    // 3) softplus(dt), decay
    dt_kernel<<<(B_ * NH_) / 256, 256, 0, stream>>>(dt_raw, a_p, dtsp, decay);

    // 4) Bandwidth-dominant SSM state update + y contraction
    ssm_update_kernel<<<B_ * NH_, 256, 0, stream>>>(ssm0, x_act, b_act, c_act,
                                                    dtsp, decay, d_p, ssm_new, y_buf);

    // 5) Gate + RMSNorm
    gate_norm_kernel<<<B_, 256, 0, stream>>>(y_buf, z_pre, nw, yn_buf);

    // 6) Output projection (fp32 WMMA)
    gemm_f32_wmma<<<HIDDEN_ / 64, 256, 0, stream>>>(yn_buf, W_out, nullptr, out, INNER_, HIDDEN_);
}